// Siamese_33895881900370
// MI455X (gfx1250) — compile-verified
//
#include <hip/hip_runtime.h>

// out[n,m] = sum_f (note[n,f]-lab[m,f])*W[f] + b
//          = (note.W)[n] + b  -  (lab.W)[m]      (rank-1 separable)
// Phase 1: row dots via V_WMMA_F32_16X16X4_F32 (exact f32, matches reference)
// Phase 2: streaming broadcast subtract, non-temporal stores (bandwidth bound)

#define N_ROWS 2048
#define M_ROWS 512
#define FDIM   256

typedef __attribute__((ext_vector_type(2))) float v2f;
typedef __attribute__((ext_vector_type(4))) float v4f;
typedef __attribute__((ext_vector_type(8))) float v8f;

// ---------------------------------------------------------------------------
// Phase 1: one wave32 per 16-row chunk. Rows [0,N) come from noteFea -> a[],
// rows [N, N+M) come from labFea -> c[]. N and M are multiples of 16, so a
// chunk never straddles the two sources.
//
// A fragment (16x4 f32, 2 VGPRs): lane L holds M = L%16, K = (L/16)*2 + {0,1}.
// B fragment (4x16 f32, 2 VGPRs): row K = (L/16)*2 + {0,1}, all 16 columns
// identical (= W[k]), so every column of the 16x16 accumulator holds the dots.
// D layout: lane 0 -> rows 0..7 in its 8 acc VGPRs, lane 16 -> rows 8..15.
// ---------------------------------------------------------------------------
__global__ __launch_bounds__(256) void siamese_dots(
    const float* __restrict__ noteFea,
    const float* __restrict__ labFea,
    const float* __restrict__ W,
    const float* __restrict__ bptr,
    float* __restrict__ a,   // [N_ROWS]  note.W + b
    float* __restrict__ c)   // [M_ROWS]  lab.W
{
  const int lane    = threadIdx.x & 31;
  const int wave    = threadIdx.x >> 5;
  const int chunk   = blockIdx.x * 8 + wave;   // 16-row chunk id
  const int rowBase = chunk * 16;
  const int half    = lane >> 4;               // 0: K=0,1  1: K=2,3
  const int r       = lane & 15;               // M within chunk
  const int rowG    = rowBase + r;

  const float* src = (rowG < N_ROWS)
                   ? (noteFea + (size_t)rowG * FDIM)
                   : (labFea  + (size_t)(rowG - N_ROWS) * FDIM);

  v8f acc0 = {};
  v8f acc1 = {};
  for (int k = 0; k < FDIM; k += 8) {
    {
      const int kk = k + half * 2;
      v2f aF = *(const v2f*)(src + kk);
      v2f bF = { W[kk], W[kk + 1] };
      acc0 = __builtin_amdgcn_wmma_f32_16x16x4_f32(
          false, aF, false, bF, (short)0, acc0, false, false);
    }
    {
      const int kk = k + 4 + half * 2;
      v2f aF = *(const v2f*)(src + kk);
      v2f bF = { W[kk], W[kk + 1] };
      acc1 = __builtin_amdgcn_wmma_f32_16x16x4_f32(
          false, aF, false, bF, (short)0, acc1, false, false);
    }
  }
  v8f acc = acc0 + acc1;   // D-layout is elementwise, so vector add is valid

  // Column 0 of D: lane 0 has rows rowBase+0..7, lane 16 has rowBase+8..15.
  if (r == 0) {
    const int   row0 = rowBase + half * 8;
    const float bias = bptr[0];
    if (row0 < N_ROWS) {
      #pragma unroll
      for (int i = 0; i < 8; ++i) a[row0 + i] = acc[i] + bias;
    } else {
      const int m0 = row0 - N_ROWS;
      #pragma unroll
      for (int i = 0; i < 8; ++i) c[m0 + i] = acc[i];
    }
  }
}

// ---------------------------------------------------------------------------
// Phase 2: out[n, m..m+3] = a[n] - c[m..m+3]. 4 MB write-once stream -> NT
// stores. a/c total 10 KB, stay resident in WGP$/L2.
// ---------------------------------------------------------------------------
__global__ __launch_bounds__(256) void siamese_outer(
    const float* __restrict__ a,
    const float* __restrict__ c,
    float* __restrict__ out)
{
  const int tid = blockIdx.x * blockDim.x + threadIdx.x;  // [0, N*M/4)
  const int n   = tid >> 7;            // M/4 = 128 float4's per row
  const int m4  = (tid & 127) << 2;

  const float an = a[n];
  v4f cv = *(const v4f*)(c + m4);
  v4f o;
  o.x = an - cv.x;
  o.y = an - cv.y;
  o.z = an - cv.z;
  o.w = an - cv.w;
  __builtin_nontemporal_store(o, (v4f*)(out + (size_t)n * M_ROWS + m4));
}

extern "C" void kernel_launch(void* const* d_in, const int* in_sizes, int n_in,
                              void* d_out, int out_size, void* d_ws, size_t ws_size,
                              hipStream_t stream) {
  const float* noteFea = (const float*)d_in[0];  // [N, F]
  const float* labFea  = (const float*)d_in[1];  // [M, F]
  const float* W       = (const float*)d_in[2];  // [F]
  const float* b       = (const float*)d_in[3];  // [1]
  float* out = (float*)d_out;                    // [N, M]

  float* a = (float*)d_ws;       // N_ROWS floats
  float* c = a + N_ROWS;         // M_ROWS floats  (10 KB total scratch)

  // 160 wave-chunks of 16 rows each; 8 waves/block -> 20 blocks.
  siamese_dots<<<(N_ROWS + M_ROWS) / (16 * 8), 256, 0, stream>>>(
      noteFea, labFea, W, b, a, c);

  // N*M/4 float4 stores, 256 threads/block -> 1024 blocks.
  siamese_outer<<<(N_ROWS * M_ROWS / 4) / 256, 256, 0, stream>>>(a, c, out);
}